// ImprovedMPS_39298950758586
// MI455X (gfx1250) — compile-verified
//
#include <hip/hip_runtime.h>

typedef float v2f __attribute__((ext_vector_type(2)));
typedef float v4f __attribute__((ext_vector_type(4)));
typedef float v8f __attribute__((ext_vector_type(8)));

#define NSITES 784
#define DBOND  64
#define KDIM   128                 // DBOND * PHYS
#define NCLS   10
#define A_SITE (KDIM * DBOND)      // 8192 floats per site
#define A_SITE_V4 (A_SITE / 4)     // 2048 float4 per site
#define PACK_STRIDE 160            // floats per packed row-pair; 160 % 64 == 32 -> conflict-free b64 reads
#define L_STRIDE 68                // padded left row stride (16B aligned, conflict-free)
#define WAVES 4
#define THREADS (WAVES * 32)
#define ROWS_PER_BLOCK (WAVES * 16)
#define V4_PER_THREAD (A_SITE_V4 / THREADS)   // 16

// Only meaningful in the device pass; host pass obviously lacks AMDGPU builtins.
#if defined(__HIP_DEVICE_COMPILE__) && !__has_builtin(__builtin_amdgcn_wmma_f32_16x16x4_f32)
#error "device pass: __builtin_amdgcn_wmma_f32_16x16x4_f32 not available"
#endif

// Per site: left_new[16x64] = left_aug[16x128] @ A_i[128x64] per wave, via
// V_WMMA_F32_16X16X4_F32 (M=16,N=16,K=4): 32 k-steps x 4 n-tiles.
// A-operand (ISA 7.12.2, 32-bit A 16x4): VGPR0 = K {0 | 2}, VGPR1 = K {1 | 3} by
// lane half; global K = 4*kk + 2*half + {0,1} -> l = 2*kk + half, p = {0,1}:
//   Aop = { left[m,l]*x0, left[m,l]*x1 }.
// B-operand mirrored (VGPR0 rows K=0/2, VGPR1 rows K=1/3) -> rows
// (4kk+2half, 4kk+2half+1), one conflict-free ds_load_b64 from the pair-packed tile.
// A-tile staging is software-pipelined through a 64-VGPR register buffer: the
// global loads for site i+1 are issued right after site i's ds-stores and their
// latency is hidden under site i's 128 WMMAs.
__global__ __launch_bounds__(THREADS)
void mps_chain_kernel(const float* __restrict__ x,
                      const float* __restrict__ A,
                      const float* __restrict__ W,
                      const float* __restrict__ bvec,
                      float* __restrict__ out)
{
  __shared__ float A_lds[64 * PACK_STRIDE];          // 40960 B, element B[k][n] at [(k>>1)*160 + 2n + (k&1)]
  __shared__ float left_lds[WAVES][16 * L_STRIDE];   // 17408 B, wave-private left state

  const int tid  = threadIdx.x;
  const int wave = tid >> 5;
  const int lane = tid & 31;
  const int half = lane >> 4;        // 0: lanes 0-15, 1: lanes 16-31
  const int mrow = lane & 15;        // M index within the wave's 16-row tile

  const int row0   = blockIdx.x * ROWS_PER_BLOCK;
  const int my_row = row0 + wave * 16 + mrow;

  // left0 = e0 (zero-padded ones vector)
  for (int j = lane; j < 16 * L_STRIDE; j += 32) left_lds[wave][j] = 0.0f;
  if (lane < 16) left_lds[wave][lane * L_STRIDE] = 1.0f;

  const float2* xrow = reinterpret_cast<const float2*>(x) + (size_t)my_row * NSITES;

  const v8f vzero = {};
  v8f C[4];

  // Register-resident staging buffer: holds the NEXT site's A tile.
  v4f rbuf[V4_PER_THREAD];
  {
    const v4f* src0 = reinterpret_cast<const v4f*>(A);
#pragma unroll
    for (int c = 0; c < V4_PER_THREAD; ++c) rbuf[c] = src0[c * THREADS + tid];
  }

  for (int site = 0; site < NSITES; ++site) {
    __syncthreads();   // all waves done reading previous A tile
    // Scatter the register-staged tile into the pair-packed LDS layout
    // (conflict-free b32 stores; compiler merges neighbors into 2addr stores).
#pragma unroll
    for (int c = 0; c < V4_PER_THREAD; ++c) {
      int v  = c * THREADS + tid;     // float4 index
      int k  = v >> 4;                // row (64 cols = 16 float4)
      int n0 = (v & 15) << 2;         // first column of this float4
      v4f d  = rbuf[c];
      float* dst = &A_lds[(k >> 1) * PACK_STRIDE + (k & 1)];
#pragma unroll
      for (int e = 0; e < 4; ++e) dst[2 * (n0 + e)] = d[e];
    }
    // Issue next site's tile loads now; latency hidden under this site's WMMAs.
    if (site + 1 < NSITES) {
      const v4f* srcn = reinterpret_cast<const v4f*>(A + (size_t)(site + 1) * A_SITE);
#pragma unroll
      for (int c = 0; c < V4_PER_THREAD; ++c) rbuf[c] = srcn[c * THREADS + tid];
    }
    float2 xv = xrow[site];             // pixel feature pair for this lane's row
    __syncthreads();   // A tile ready

    C[0] = vzero; C[1] = vzero; C[2] = vzero; C[3] = vzero;
    const float* lrow = &left_lds[wave][mrow * L_STRIDE + half];
#pragma unroll
    for (int kk = 0; kk < 32; ++kk) {
      float lv = lrow[2 * kk];                         // left[m, 2kk+half]
      v2f Aop = { lv * xv.x, lv * xv.y };              // fold x into left_aug
      const float* brow = &A_lds[(2 * kk + half) * PACK_STRIDE + 2 * mrow];
#pragma unroll
      for (int t = 0; t < 4; ++t) {
        v2f Bop = *reinterpret_cast<const v2f*>(brow + 32 * t);  // one b64, conflict-free
        C[t] = __builtin_amdgcn_wmma_f32_16x16x4_f32(false, Aop, false, Bop,
                                                     (short)0, C[t], false, false);
      }
    }
    // C layout (VGPR j: M = j + 8*half, N = t*16 + lane%16) -> left_lds row-major.
#pragma unroll
    for (int t = 0; t < 4; ++t)
#pragma unroll
      for (int j = 0; j < 8; ++j)
        left_lds[wave][(j + 8 * half) * L_STRIDE + t * 16 + mrow] = C[t][j];
  }

  __syncthreads();
  // Final right bond dim == 1: logits[r,c] = left[r,0]*W[c] + b[c]
  if (tid < ROWS_PER_BLOCK) {
    float lv = left_lds[tid >> 4][(tid & 15) * L_STRIDE];
    float* orow = out + (size_t)(row0 + tid) * NCLS;
#pragma unroll
    for (int c = 0; c < NCLS; ++c) orow[c] = fmaf(lv, W[c], bvec[c]);
  }
}

extern "C" void kernel_launch(void* const* d_in, const int* in_sizes, int n_in,
                              void* d_out, int out_size, void* d_ws, size_t ws_size,
                              hipStream_t stream) {
  const float* x = (const float*)d_in[0];   // [4096, 784, 2] f32
  const float* A = (const float*)d_in[1];   // [784, 64, 2, 64] f32
  const float* W = (const float*)d_in[2];   // [1, 10] f32
  const float* b = (const float*)d_in[3];   // [10] f32
  float* out = (float*)d_out;               // [4096, 10] f32

  const int batch = in_sizes[0] / (NSITES * 2);   // 4096
  dim3 grid(batch / ROWS_PER_BLOCK);              // 64 workgroups
  dim3 block(THREADS);                            // 4 waves (wave32)
  hipLaunchKernelGGL(mps_chain_kernel, grid, block, 0, stream, x, A, W, b, out);
}